// HFCFilter_59450937311643
// MI455X (gfx1250) — compile-verified
//
#include <hip/hip_runtime.h>

// Problem constants (from reference): x[32,3,512,512] f32, mask[32,1,512,512] f32
#define BB 32
#define CC 3
#define HW 262144               // 512*512
#define NG 96                   // BB*CC groups
#define SPLIT 8                 // sub-blocks per group for streaming passes
#define CHUNK (HW / SPLIT)      // 32768 elements per sub-block

typedef unsigned int uint32;
typedef __attribute__((ext_vector_type(2))) float v2f;
typedef __attribute__((ext_vector_type(4))) float v4f;
typedef __attribute__((ext_vector_type(8))) float v8f;

#if defined(__gfx1250__) && __has_builtin(__builtin_amdgcn_wmma_f32_16x16x4_f32)
#define HAVE_WMMA_F32X4 1
#endif

// ===========================================================================
// Shared helpers
// ===========================================================================

__global__ void zero_u32(uint32* __restrict__ p, int n) {
    int i = (int)blockIdx.x * (int)blockDim.x + (int)threadIdx.x;
    if (i < n) p[i] = 0u;
}

// Block-parallel order-statistic selection over h[256*nb] (contiguous bins per
// thread). 256 threads: per-thread partial sums -> Hillis-Steele inclusive
// scan in LDS -> the single owning thread walks <= nb bins. Results to shared
// scalars. Serial depth ~ nb + log2(256).
__device__ __forceinline__ void select_bin(const uint32* __restrict__ h, int nb,
                                           uint32 r, uint32* sc,
                                           uint32* out_bin, uint32* out_rem) {
    const int t = (int)threadIdx.x;
    const int base = t * nb;
    uint32 loc[8];
    uint32 p = 0;
    for (int e = 0; e < nb; ++e) { loc[e] = h[base + e]; p += loc[e]; }
    sc[t] = p;
    __syncthreads();
    for (int off = 1; off < 256; off <<= 1) {
        uint32 v = (t >= off) ? sc[t - off] : 0u;
        __syncthreads();
        sc[t] += v;
        __syncthreads();
    }
    uint32 incl = sc[t];
    uint32 excl = incl - p;
    if (r >= excl && r < incl) {
        uint32 cum = excl;
        for (int e = 0; e < nb; ++e) {
            uint32 c = loc[e];
            if (cum + c > r) { *out_bin = (uint32)(base + e); *out_rem = r - cum; break; }
            cum += c;
        }
    }
    __syncthreads();
}

// ===========================================================================
// SPLIT path: exact median via 3-pass radix-histogram selection on f32 keys
// (inputs are non-negative so u32 bit order == value order). Each pass:
// grid(NG, SPLIT) blocks build LDS histograms and flush to per-group global
// histograms; a small 256-thread scan kernel does the parallel rank select.
// ===========================================================================

__global__ void p1_hist(const float* __restrict__ x, uint32* __restrict__ gh) {
    __shared__ uint32 hist[2048];
    const int g = blockIdx.x;
    const float* xg = x + (size_t)g * HW + (size_t)blockIdx.y * CHUNK;
    for (int i = threadIdx.x; i < 2048; i += blockDim.x) hist[i] = 0u;
    __syncthreads();
    for (int i = threadIdx.x * 4; i < CHUNK; i += blockDim.x * 4) {
        __builtin_prefetch(xg + i + 8192, 0, 0);      // global_prefetch_b8
        v4f v = *(const v4f*)(xg + i);
        atomicAdd(&hist[__float_as_uint(v.x) >> 21], 1u);
        atomicAdd(&hist[__float_as_uint(v.y) >> 21], 1u);
        atomicAdd(&hist[__float_as_uint(v.z) >> 21], 1u);
        atomicAdd(&hist[__float_as_uint(v.w) >> 21], 1u);
    }
    __syncthreads();
    for (int i = threadIdx.x; i < 2048; i += blockDim.x) {
        uint32 c = hist[i];
        if (c) atomicAdd(&gh[(size_t)g * 2048 + i], c);
    }
}

__global__ void p1_scan(const uint32* __restrict__ gh,
                        uint32* __restrict__ path_a, uint32* __restrict__ rank_a,
                        uint32* __restrict__ path_b, uint32* __restrict__ rank_b) {
    __shared__ uint32 sc[256];
    __shared__ uint32 sbin, srem;
    const int g = blockIdx.x;
    const uint32* h = gh + (size_t)g * 2048;
    select_bin(h, 8, HW / 2 - 1, sc, &sbin, &srem);
    if (threadIdx.x == 0) { path_a[g] = sbin << 21; rank_a[g] = srem; }
    __syncthreads();
    select_bin(h, 8, HW / 2, sc, &sbin, &srem);
    if (threadIdx.x == 0) { path_b[g] = sbin << 21; rank_b[g] = srem; }
}

__global__ void p2_hist(const float* __restrict__ x,
                        uint32* __restrict__ ghA, uint32* __restrict__ ghB,
                        const uint32* __restrict__ path_a, const uint32* __restrict__ path_b) {
    __shared__ uint32 ha[2048];
    __shared__ uint32 hb[2048];
    const int g = blockIdx.x;
    const float* xg = x + (size_t)g * HW + (size_t)blockIdx.y * CHUNK;
    const uint32 pa = path_a[g], pb = path_b[g];
    for (int i = threadIdx.x; i < 2048; i += blockDim.x) { ha[i] = 0u; hb[i] = 0u; }
    __syncthreads();
    for (int i = threadIdx.x * 4; i < CHUNK; i += blockDim.x * 4) {
        __builtin_prefetch(xg + i + 8192, 0, 0);
        v4f v = *(const v4f*)(xg + i);
        uint32 k0 = __float_as_uint(v.x), k1 = __float_as_uint(v.y);
        uint32 k2 = __float_as_uint(v.z), k3 = __float_as_uint(v.w);
        if ((k0 & 0xFFE00000u) == pa) atomicAdd(&ha[(k0 >> 10) & 2047u], 1u);
        if ((k1 & 0xFFE00000u) == pa) atomicAdd(&ha[(k1 >> 10) & 2047u], 1u);
        if ((k2 & 0xFFE00000u) == pa) atomicAdd(&ha[(k2 >> 10) & 2047u], 1u);
        if ((k3 & 0xFFE00000u) == pa) atomicAdd(&ha[(k3 >> 10) & 2047u], 1u);
        if ((k0 & 0xFFE00000u) == pb) atomicAdd(&hb[(k0 >> 10) & 2047u], 1u);
        if ((k1 & 0xFFE00000u) == pb) atomicAdd(&hb[(k1 >> 10) & 2047u], 1u);
        if ((k2 & 0xFFE00000u) == pb) atomicAdd(&hb[(k2 >> 10) & 2047u], 1u);
        if ((k3 & 0xFFE00000u) == pb) atomicAdd(&hb[(k3 >> 10) & 2047u], 1u);
    }
    __syncthreads();
    for (int i = threadIdx.x; i < 2048; i += blockDim.x) {
        uint32 ca = ha[i]; if (ca) atomicAdd(&ghA[(size_t)g * 2048 + i], ca);
        uint32 cb = hb[i]; if (cb) atomicAdd(&ghB[(size_t)g * 2048 + i], cb);
    }
}

__global__ void p2_scan(const uint32* __restrict__ ghA, const uint32* __restrict__ ghB,
                        uint32* __restrict__ path_a, uint32* __restrict__ rank_a,
                        uint32* __restrict__ path_b, uint32* __restrict__ rank_b) {
    __shared__ uint32 sc[256];
    __shared__ uint32 sbin, srem;
    const int g = blockIdx.x;
    const uint32 pa = path_a[g], ra = rank_a[g];
    const uint32 pb = path_b[g], rb = rank_b[g];
    select_bin(ghA + (size_t)g * 2048, 8, ra, sc, &sbin, &srem);
    if (threadIdx.x == 0) { path_a[g] = pa | (sbin << 10); rank_a[g] = srem; }
    __syncthreads();
    select_bin(ghB + (size_t)g * 2048, 8, rb, sc, &sbin, &srem);
    if (threadIdx.x == 0) { path_b[g] = pb | (sbin << 10); rank_b[g] = srem; }
}

__global__ void p3_hist(const float* __restrict__ x,
                        uint32* __restrict__ ghA, uint32* __restrict__ ghB,
                        const uint32* __restrict__ path_a, const uint32* __restrict__ path_b) {
    __shared__ uint32 ha[1024];
    __shared__ uint32 hb[1024];
    const int g = blockIdx.x;
    const float* xg = x + (size_t)g * HW + (size_t)blockIdx.y * CHUNK;
    const uint32 pa = path_a[g], pb = path_b[g];
    for (int i = threadIdx.x; i < 1024; i += blockDim.x) { ha[i] = 0u; hb[i] = 0u; }
    __syncthreads();
    for (int i = threadIdx.x * 4; i < CHUNK; i += blockDim.x * 4) {
        __builtin_prefetch(xg + i + 8192, 0, 0);
        v4f v = *(const v4f*)(xg + i);
        uint32 k0 = __float_as_uint(v.x), k1 = __float_as_uint(v.y);
        uint32 k2 = __float_as_uint(v.z), k3 = __float_as_uint(v.w);
        if ((k0 & 0xFFFFFC00u) == pa) atomicAdd(&ha[k0 & 1023u], 1u);
        if ((k1 & 0xFFFFFC00u) == pa) atomicAdd(&ha[k1 & 1023u], 1u);
        if ((k2 & 0xFFFFFC00u) == pa) atomicAdd(&ha[k2 & 1023u], 1u);
        if ((k3 & 0xFFFFFC00u) == pa) atomicAdd(&ha[k3 & 1023u], 1u);
        if ((k0 & 0xFFFFFC00u) == pb) atomicAdd(&hb[k0 & 1023u], 1u);
        if ((k1 & 0xFFFFFC00u) == pb) atomicAdd(&hb[k1 & 1023u], 1u);
        if ((k2 & 0xFFFFFC00u) == pb) atomicAdd(&hb[k2 & 1023u], 1u);
        if ((k3 & 0xFFFFFC00u) == pb) atomicAdd(&hb[k3 & 1023u], 1u);
    }
    __syncthreads();
    for (int i = threadIdx.x; i < 1024; i += blockDim.x) {
        uint32 ca = ha[i]; if (ca) atomicAdd(&ghA[(size_t)g * 1024 + i], ca);
        uint32 cb = hb[i]; if (cb) atomicAdd(&ghB[(size_t)g * 1024 + i], cb);
    }
}

__global__ void p3_scan(const uint32* __restrict__ ghA, const uint32* __restrict__ ghB,
                        const uint32* __restrict__ path_a, const uint32* __restrict__ rank_a,
                        const uint32* __restrict__ path_b, const uint32* __restrict__ rank_b,
                        float* __restrict__ med) {
    __shared__ uint32 sc[256];
    __shared__ uint32 sbin, srem;
    const int g = blockIdx.x;
    const uint32 pa = path_a[g], ra = rank_a[g];
    const uint32 pb = path_b[g], rb = rank_b[g];
    select_bin(ghA + (size_t)g * 1024, 4, ra, sc, &sbin, &srem);
    uint32 keyA = pa | sbin;             // valid in thread 0 after barrier below
    __syncthreads();
    select_bin(ghB + (size_t)g * 1024, 4, rb, sc, &sbin, &srem);
    if (threadIdx.x == 0) {
        uint32 keyB = pb | sbin;
        med[g] = 0.5f * (__uint_as_float(keyA) + __uint_as_float(keyB)) + 0.2f;
    }
}

// ===========================================================================
// Quantile histogram (split): temp = trunc(y*256)/256 lives on a 1/256 grid,
// so a 512-bin integer histogram is an exact order-statistic oracle.
// ===========================================================================

__global__ void q_hist(const float* __restrict__ x, const float* __restrict__ mask,
                       const float* __restrict__ med, uint32* __restrict__ gh) {
    __shared__ uint32 hq[512];
    const int g = blockIdx.x;
    const int b = g / CC;
    const float* xg = x + (size_t)g * HW + (size_t)blockIdx.y * CHUNK;
    const float* mg = mask + (size_t)b * HW + (size_t)blockIdx.y * CHUNK;
    const float fill = med[g];
    for (int i = threadIdx.x; i < 512; i += blockDim.x) hq[i] = 0u;
    __syncthreads();
    for (int i = threadIdx.x * 4; i < CHUNK; i += blockDim.x * 4) {
        __builtin_prefetch(xg + i + 8192, 0, 0);
        __builtin_prefetch(mg + i + 8192, 0, 0);
        v4f xv = *(const v4f*)(xg + i);
        v4f mv = *(const v4f*)(mg + i);
        float y0 = mv.x * xv.x + (1.0f - mv.x) * fill;
        float y1 = mv.y * xv.y + (1.0f - mv.y) * fill;
        float y2 = mv.z * xv.z + (1.0f - mv.z) * fill;
        float y3 = mv.w * xv.w + (1.0f - mv.w) * fill;
        int b0 = min(max((int)(y0 * 256.0f), 0), 511);
        int b1 = min(max((int)(y1 * 256.0f), 0), 511);
        int b2 = min(max((int)(y2 * 256.0f), 0), 511);
        int b3 = min(max((int)(y3 * 256.0f), 0), 511);
        atomicAdd(&hq[b0], 1u); atomicAdd(&hq[b1], 1u);
        atomicAdd(&hq[b2], 1u); atomicAdd(&hq[b3], 1u);
    }
    __syncthreads();
    for (int i = threadIdx.x; i < 512; i += blockDim.x) {
        uint32 c = hq[i];
        if (c) atomicAdd(&gh[(size_t)g * 512 + i], c);
    }
}

// Percentile extraction. WMMA f32 16x16x4 with B = all-ones computes row sums
// of 16x4 histogram chunks -> 128 coarse 4-bin block sums (exact: counts<2^24).
// Then a 128-wide parallel scan + binary search per endpoint.
__device__ __forceinline__ float nth_from_cdf(const float* ccdf, const uint32* hq, uint32 k) {
    int lo = 0, hi = 127;
    while (lo < hi) {
        int mid = (lo + hi) >> 1;
        if ((uint32)ccdf[mid] > k) hi = mid; else lo = mid + 1;
    }
    uint32 cum = (lo > 0) ? (uint32)ccdf[lo - 1] : 0u;
    int bin = lo * 4;
    for (int e = 0; e < 3; ++e) {
        uint32 c = hq[bin];
        if (cum + c > k) break;
        cum += c; ++bin;
    }
    return (float)bin * (1.0f / 256.0f);
}

__global__ void q_scan(const uint32* __restrict__ gh,
                       float* __restrict__ lo_out, float* __restrict__ scale_out) {
    __shared__ uint32 hq[512];
    __shared__ float coarse[128];
    __shared__ float ccdf[128];
    const int g = blockIdx.x;
    const int t = (int)threadIdx.x;
    const uint32* h = gh + (size_t)g * 512;
    hq[t] = h[t];
    hq[t + 256] = h[t + 256];
    __syncthreads();
#if __has_builtin(__builtin_amdgcn_s_wait_tensorcnt)
    __builtin_amdgcn_s_wait_tensorcnt(0);
#endif
    {
        // Every lane of every wave runs the WMMA (EXEC all-ones requirement);
        // only wave-0 lanes 0/16 commit results (redundant writes identical).
        const int L  = t & 31;
        const int m  = L & 15;
        const int k0 = (L < 16) ? 0 : 2;
#ifdef HAVE_WMMA_F32X4
        v2f ones; ones.x = 1.0f; ones.y = 1.0f;
        for (int j = 0; j < 8; ++j) {
            v2f a;
            a.x = (float)hq[j * 64 + m * 4 + k0];
            a.y = (float)hq[j * 64 + m * 4 + k0 + 1];
            v8f c = {};
            v8f d = __builtin_amdgcn_wmma_f32_16x16x4_f32(
                false, a, false, ones, (short)0, c, false, false);
            if (t == 0)  { for (int e = 0; e < 8; ++e) coarse[j * 16 + e]     = d[e]; }
            if (t == 16) { for (int e = 0; e < 8; ++e) coarse[j * 16 + 8 + e] = d[e]; }
        }
#else
        if (t < 128)
            coarse[t] = (float)(hq[4 * t] + hq[4 * t + 1] + hq[4 * t + 2] + hq[4 * t + 3]);
#endif
    }
    __syncthreads();
    if (t < 128) ccdf[t] = coarse[t];
    __syncthreads();
    for (int off = 1; off < 128; off <<= 1) {
        float v = 0.0f;
        if (t < 128 && t >= off) v = ccdf[t - off];
        __syncthreads();
        if (t < 128) ccdf[t] += v;
        __syncthreads();
    }
    if (t == 0) {
        // jnp.percentile, method="linear": pos = q/100 * (N-1)
        double pos_lo = 0.03 * (double)(HW - 1);
        double pos_hi = 0.97 * (double)(HW - 1);
        uint32 kl = (uint32)pos_lo;  double fl = pos_lo - (double)kl;
        uint32 kh = (uint32)pos_hi;  double fh = pos_hi - (double)kh;
        float l0 = nth_from_cdf(ccdf, hq, kl);
        float l1 = nth_from_cdf(ccdf, hq, kl + 1);
        float h0 = nth_from_cdf(ccdf, hq, kh);
        float h1 = nth_from_cdf(ccdf, hq, kh + 1);
        double lov = (double)l0 + fl * ((double)l1 - (double)l0);
        double hiv = (double)h0 + fh * ((double)h1 - (double)h0);
        lo_out[g]    = (float)lov;
        scale_out[g] = (float)(1.0 / (hiv - lov));
    }
}

// ===========================================================================
// Final pointwise normalize: out = ((mask*x + (1-mask)*med) - lo)*scale*mask
// Streaming float4; nontemporal stores (output has no reuse, keep x/mask in L2).
// ===========================================================================
__global__ void finalize_kernel(const float* __restrict__ x, const float* __restrict__ mask,
                                const float* __restrict__ med, const float* __restrict__ lo,
                                const float* __restrict__ scale, float* __restrict__ out) {
    const int g = blockIdx.y;
    const int b = g / CC;
    const size_t base  = (size_t)g * HW;
    const size_t mbase = (size_t)b * HW;
    const int i = ((int)blockIdx.x * (int)blockDim.x + (int)threadIdx.x) * 4;
    const float fill = med[g];
    const float l = lo[g];
    const float s = scale[g];
    __builtin_prefetch(x + base + i + 4096, 0, 0);
    __builtin_prefetch(mask + mbase + i + 4096, 0, 0);
    v4f xv = *(const v4f*)(x + base + i);
    v4f mv = *(const v4f*)(mask + mbase + i);
    v4f o;
    o.x = ((mv.x * xv.x + (1.0f - mv.x) * fill) - l) * s * mv.x;
    o.y = ((mv.y * xv.y + (1.0f - mv.y) * fill) - l) * s * mv.y;
    o.z = ((mv.z * xv.z + (1.0f - mv.z) * fill) - l) * s * mv.z;
    o.w = ((mv.w * xv.w + (1.0f - mv.w) * fill) - l) * s * mv.w;
    __builtin_nontemporal_store(o, (v4f*)(out + base + i));
}

// ===========================================================================
// Fallback (monolithic, one workgroup per group) — used if ws_size is small.
// ===========================================================================

__global__ void median_mono1(const float* __restrict__ x,
                             uint32* __restrict__ path_a, uint32* __restrict__ rank_a,
                             uint32* __restrict__ path_b, uint32* __restrict__ rank_b) {
    __shared__ uint32 hist[2048];
    __shared__ uint32 sc[256];
    __shared__ uint32 sbin, srem;
    const int g = blockIdx.x;
    const float* xg = x + (size_t)g * HW;
    for (int i = threadIdx.x; i < 2048; i += blockDim.x) hist[i] = 0u;
    __syncthreads();
    for (int i = threadIdx.x * 4; i < HW; i += blockDim.x * 4) {
        v4f v = *(const v4f*)(xg + i);
        atomicAdd(&hist[__float_as_uint(v.x) >> 21], 1u);
        atomicAdd(&hist[__float_as_uint(v.y) >> 21], 1u);
        atomicAdd(&hist[__float_as_uint(v.z) >> 21], 1u);
        atomicAdd(&hist[__float_as_uint(v.w) >> 21], 1u);
    }
    __syncthreads();
    if (threadIdx.x < 256) select_bin(hist, 8, HW / 2 - 1, sc, &sbin, &srem);
    __syncthreads();
    if (threadIdx.x == 0) { path_a[g] = sbin << 21; rank_a[g] = srem; }
    __syncthreads();
    if (threadIdx.x < 256) select_bin(hist, 8, HW / 2, sc, &sbin, &srem);
    __syncthreads();
    if (threadIdx.x == 0) { path_b[g] = sbin << 21; rank_b[g] = srem; }
}

__global__ void median_mono2(const float* __restrict__ x,
                             uint32* __restrict__ path_a, uint32* __restrict__ rank_a,
                             uint32* __restrict__ path_b, uint32* __restrict__ rank_b) {
    __shared__ uint32 ha[2048];
    __shared__ uint32 hb[2048];
    __shared__ uint32 sc[256];
    __shared__ uint32 sbin, srem;
    const int g = blockIdx.x;
    const float* xg = x + (size_t)g * HW;
    const uint32 pa = path_a[g], pb = path_b[g];
    const uint32 ra = rank_a[g], rb = rank_b[g];
    for (int i = threadIdx.x; i < 2048; i += blockDim.x) { ha[i] = 0u; hb[i] = 0u; }
    __syncthreads();
    for (int i = threadIdx.x * 4; i < HW; i += blockDim.x * 4) {
        v4f v = *(const v4f*)(xg + i);
        uint32 k0 = __float_as_uint(v.x), k1 = __float_as_uint(v.y);
        uint32 k2 = __float_as_uint(v.z), k3 = __float_as_uint(v.w);
        if ((k0 & 0xFFE00000u) == pa) atomicAdd(&ha[(k0 >> 10) & 2047u], 1u);
        if ((k1 & 0xFFE00000u) == pa) atomicAdd(&ha[(k1 >> 10) & 2047u], 1u);
        if ((k2 & 0xFFE00000u) == pa) atomicAdd(&ha[(k2 >> 10) & 2047u], 1u);
        if ((k3 & 0xFFE00000u) == pa) atomicAdd(&ha[(k3 >> 10) & 2047u], 1u);
        if ((k0 & 0xFFE00000u) == pb) atomicAdd(&hb[(k0 >> 10) & 2047u], 1u);
        if ((k1 & 0xFFE00000u) == pb) atomicAdd(&hb[(k1 >> 10) & 2047u], 1u);
        if ((k2 & 0xFFE00000u) == pb) atomicAdd(&hb[(k2 >> 10) & 2047u], 1u);
        if ((k3 & 0xFFE00000u) == pb) atomicAdd(&hb[(k3 >> 10) & 2047u], 1u);
    }
    __syncthreads();
    if (threadIdx.x < 256) select_bin(ha, 8, ra, sc, &sbin, &srem);
    __syncthreads();
    if (threadIdx.x == 0) { path_a[g] = pa | (sbin << 10); rank_a[g] = srem; }
    __syncthreads();
    if (threadIdx.x < 256) select_bin(hb, 8, rb, sc, &sbin, &srem);
    __syncthreads();
    if (threadIdx.x == 0) { path_b[g] = pb | (sbin << 10); rank_b[g] = srem; }
}

__global__ void median_mono3(const float* __restrict__ x,
                             const uint32* __restrict__ path_a, const uint32* __restrict__ rank_a,
                             const uint32* __restrict__ path_b, const uint32* __restrict__ rank_b,
                             float* __restrict__ med) {
    __shared__ uint32 ha[1024];
    __shared__ uint32 hb[1024];
    __shared__ uint32 sc[256];
    __shared__ uint32 sbin, srem;
    const int g = blockIdx.x;
    const float* xg = x + (size_t)g * HW;
    const uint32 pa = path_a[g], pb = path_b[g];
    const uint32 ra = rank_a[g], rb = rank_b[g];
    for (int i = threadIdx.x; i < 1024; i += blockDim.x) { ha[i] = 0u; hb[i] = 0u; }
    __syncthreads();
    for (int i = threadIdx.x * 4; i < HW; i += blockDim.x * 4) {
        v4f v = *(const v4f*)(xg + i);
        uint32 k0 = __float_as_uint(v.x), k1 = __float_as_uint(v.y);
        uint32 k2 = __float_as_uint(v.z), k3 = __float_as_uint(v.w);
        if ((k0 & 0xFFFFFC00u) == pa) atomicAdd(&ha[k0 & 1023u], 1u);
        if ((k1 & 0xFFFFFC00u) == pa) atomicAdd(&ha[k1 & 1023u], 1u);
        if ((k2 & 0xFFFFFC00u) == pa) atomicAdd(&ha[k2 & 1023u], 1u);
        if ((k3 & 0xFFFFFC00u) == pa) atomicAdd(&ha[k3 & 1023u], 1u);
        if ((k0 & 0xFFFFFC00u) == pb) atomicAdd(&hb[k0 & 1023u], 1u);
        if ((k1 & 0xFFFFFC00u) == pb) atomicAdd(&hb[k1 & 1023u], 1u);
        if ((k2 & 0xFFFFFC00u) == pb) atomicAdd(&hb[k2 & 1023u], 1u);
        if ((k3 & 0xFFFFFC00u) == pb) atomicAdd(&hb[k3 & 1023u], 1u);
    }
    __syncthreads();
    uint32 keyA = 0;
    if (threadIdx.x < 256) select_bin(ha, 4, ra, sc, &sbin, &srem);
    __syncthreads();
    keyA = pa | sbin;
    __syncthreads();
    if (threadIdx.x < 256) select_bin(hb, 4, rb, sc, &sbin, &srem);
    __syncthreads();
    if (threadIdx.x == 0) {
        uint32 keyB = pb | sbin;
        med[g] = 0.5f * (__uint_as_float(keyA) + __uint_as_float(keyB)) + 0.2f;
    }
}

__global__ void quant_mono(const float* __restrict__ x, const float* __restrict__ mask,
                           const float* __restrict__ med,
                           float* __restrict__ lo_out, float* __restrict__ scale_out) {
    __shared__ uint32 hq[512];
    __shared__ float coarse[128];
    __shared__ float ccdf[128];
    const int g = blockIdx.x;
    const int b = g / CC;
    const float* xg = x + (size_t)g * HW;
    const float* mg = mask + (size_t)b * HW;
    const float fill = med[g];
    const int t = (int)threadIdx.x;
    for (int i = t; i < 512; i += blockDim.x) hq[i] = 0u;
    __syncthreads();
    for (int i = t * 4; i < HW; i += blockDim.x * 4) {
        v4f xv = *(const v4f*)(xg + i);
        v4f mv = *(const v4f*)(mg + i);
        float y0 = mv.x * xv.x + (1.0f - mv.x) * fill;
        float y1 = mv.y * xv.y + (1.0f - mv.y) * fill;
        float y2 = mv.z * xv.z + (1.0f - mv.z) * fill;
        float y3 = mv.w * xv.w + (1.0f - mv.w) * fill;
        atomicAdd(&hq[min(max((int)(y0 * 256.0f), 0), 511)], 1u);
        atomicAdd(&hq[min(max((int)(y1 * 256.0f), 0), 511)], 1u);
        atomicAdd(&hq[min(max((int)(y2 * 256.0f), 0), 511)], 1u);
        atomicAdd(&hq[min(max((int)(y3 * 256.0f), 0), 511)], 1u);
    }
    __syncthreads();
    {
        const int L  = t & 31;
        const int m  = L & 15;
        const int k0 = (L < 16) ? 0 : 2;
#ifdef HAVE_WMMA_F32X4
        v2f ones; ones.x = 1.0f; ones.y = 1.0f;
        for (int j = 0; j < 8; ++j) {
            v2f a;
            a.x = (float)hq[j * 64 + m * 4 + k0];
            a.y = (float)hq[j * 64 + m * 4 + k0 + 1];
            v8f c = {};
            v8f d = __builtin_amdgcn_wmma_f32_16x16x4_f32(
                false, a, false, ones, (short)0, c, false, false);
            if (t == 0)  { for (int e = 0; e < 8; ++e) coarse[j * 16 + e]     = d[e]; }
            if (t == 16) { for (int e = 0; e < 8; ++e) coarse[j * 16 + 8 + e] = d[e]; }
        }
#else
        if (t < 128)
            coarse[t] = (float)(hq[4 * t] + hq[4 * t + 1] + hq[4 * t + 2] + hq[4 * t + 3]);
#endif
    }
    __syncthreads();
    if (t < 128) ccdf[t] = coarse[t];
    __syncthreads();
    for (int off = 1; off < 128; off <<= 1) {
        float v = 0.0f;
        if (t < 128 && t >= off) v = ccdf[t - off];
        __syncthreads();
        if (t < 128) ccdf[t] += v;
        __syncthreads();
    }
    if (t == 0) {
        double pos_lo = 0.03 * (double)(HW - 1);
        double pos_hi = 0.97 * (double)(HW - 1);
        uint32 kl = (uint32)pos_lo;  double fl = pos_lo - (double)kl;
        uint32 kh = (uint32)pos_hi;  double fh = pos_hi - (double)kh;
        float l0 = nth_from_cdf(ccdf, hq, kl);
        float l1 = nth_from_cdf(ccdf, hq, kl + 1);
        float h0 = nth_from_cdf(ccdf, hq, kh);
        float h1 = nth_from_cdf(ccdf, hq, kh + 1);
        double lov = (double)l0 + fl * ((double)l1 - (double)l0);
        double hiv = (double)h0 + fh * ((double)h1 - (double)h0);
        lo_out[g]    = (float)lov;
        scale_out[g] = (float)(1.0 / (hiv - lov));
    }
}

// ===========================================================================
// Launcher. Workspace layout:
//   [0,384)      med[96] f32        [512,896)  lo[96]       [1024,..) scale[96]
//   [1536..3456) path_a/rank_a/path_b/rank_b u32[96] each (512B stride)
//   [4096, +768K)   ghA  (96*2048 u32)
//   [+768K, +1.5M)  ghB  (96*2048 u32)
// ===========================================================================
extern "C" void kernel_launch(void* const* d_in, const int* in_sizes, int n_in,
                              void* d_out, int out_size, void* d_ws, size_t ws_size,
                              hipStream_t stream) {
    const float* x    = (const float*)d_in[0];
    const float* mask = (const float*)d_in[1];
    float* out        = (float*)d_out;
    char* ws          = (char*)d_ws;

    float*  med    = (float*)(ws + 0);
    float*  lo     = (float*)(ws + 512);
    float*  scale  = (float*)(ws + 1024);
    uint32* path_a = (uint32*)(ws + 1536);
    uint32* rank_a = (uint32*)(ws + 2048);
    uint32* path_b = (uint32*)(ws + 2560);
    uint32* rank_b = (uint32*)(ws + 3072);
    uint32* ghA    = (uint32*)(ws + 4096);
    uint32* ghB    = ghA + (size_t)NG * 2048;

    (void)in_sizes; (void)n_in; (void)out_size;

    const size_t need = 4096 + (size_t)2 * NG * 2048 * sizeof(uint32);

    if (ws_size >= need) {
        const int nz1 = NG * 2048;          // ghA only
        const int nz2 = 2 * NG * 2048;      // ghA + ghB (contiguous)
        const int nzq = NG * 512;

        zero_u32<<<(nz1 + 255) / 256, 256, 0, stream>>>(ghA, nz1);
        p1_hist<<<dim3(NG, SPLIT), 1024, 0, stream>>>(x, ghA);
        p1_scan<<<NG, 256, 0, stream>>>(ghA, path_a, rank_a, path_b, rank_b);

        zero_u32<<<(nz2 + 255) / 256, 256, 0, stream>>>(ghA, nz2);
        p2_hist<<<dim3(NG, SPLIT), 1024, 0, stream>>>(x, ghA, ghB, path_a, path_b);
        p2_scan<<<NG, 256, 0, stream>>>(ghA, ghB, path_a, rank_a, path_b, rank_b);

        zero_u32<<<(nz2 + 255) / 256, 256, 0, stream>>>(ghA, nz2);
        p3_hist<<<dim3(NG, SPLIT), 1024, 0, stream>>>(x, ghA, ghB, path_a, path_b);
        p3_scan<<<NG, 256, 0, stream>>>(ghA, ghB, path_a, rank_a, path_b, rank_b, med);

        zero_u32<<<(nzq + 255) / 256, 256, 0, stream>>>(ghA, nzq);
        q_hist<<<dim3(NG, SPLIT), 1024, 0, stream>>>(x, mask, med, ghA);
        q_scan<<<NG, 256, 0, stream>>>(ghA, lo, scale);
    } else {
        median_mono1<<<NG, 1024, 0, stream>>>(x, path_a, rank_a, path_b, rank_b);
        median_mono2<<<NG, 1024, 0, stream>>>(x, path_a, rank_a, path_b, rank_b);
        median_mono3<<<NG, 1024, 0, stream>>>(x, path_a, rank_a, path_b, rank_b, med);
        quant_mono<<<NG, 1024, 0, stream>>>(x, mask, med, lo, scale);
    }

    dim3 grid(HW / (256 * 4), NG);
    finalize_kernel<<<grid, 256, 0, stream>>>(x, mask, med, lo, scale, out);
}